// MessagePassing_55405078118496
// MI455X (gfx1250) — compile-verified
//
#include <hip/hip_runtime.h>

typedef __attribute__((ext_vector_type(2))) float v2f;
typedef __attribute__((ext_vector_type(8))) float v8f;

#define N_NODES  100000
#define N_EDGES  500000
#define NODE_DIM 32

// ---------------------------------------------------------------------------
// Kernel 1: zero the accumulator (d_out) and per-node edge counts (d_ws).
// ---------------------------------------------------------------------------
__global__ void MP_zero_kernel(float* __restrict__ out, float* __restrict__ cnt) {
    int i = blockIdx.x * blockDim.x + threadIdx.x;
    if (i < N_NODES * NODE_DIM) out[i] = 0.0f;
    if (i < N_NODES)            cnt[i] = 0.0f;
}

// ---------------------------------------------------------------------------
// Kernel 2: one wave32 per edge.
//   msg = x_src (1x32) @ A_e (32x32) computed with V_WMMA_F32_16X16X4_F32:
//   - A operand (16x4): x values broadcast down all 16 M rows
//       VGPR0 lane L = x[4kb + (L<16 ? 0 : 2)], VGPR1 = x[4kb + (L<16 ? 1 : 3)]
//   - B operand (4x16): rows 4kb..4kb+3 of A_e, feature half h
//       VGPR0 lane L = A_e[4kb + (L<16?0:2)][16h + (L&15)]
//       VGPR1 lane L = A_e[4kb + (L<16?1:3)][16h + (L&15)]
//   - C (16x16): all rows identical = msg[16h + n]; after the 8-step K chain,
//     lane L holds its own feature: h = (L>=16), n = L&15  ->  feature == L.
//   Scatter: global_atomic_add_f32 into accum[dst*32 + lane].
// ---------------------------------------------------------------------------
__global__ __launch_bounds__(256) void MP_edge_kernel(
        const float* __restrict__ node_states,
        const int*   __restrict__ edge_index,
        const float* __restrict__ a_in,
        float*       __restrict__ accum,
        float*       __restrict__ cnt) {
    const int lane = threadIdx.x & 31;
    const int wave = threadIdx.x >> 5;
    const int e    = blockIdx.x * (blockDim.x >> 5) + wave;
    if (e >= N_EDGES) return;               // whole-wave guard: EXEC all-1 for WMMA

    const int src = edge_index[2 * e + 0];
    const int dst = edge_index[2 * e + 1];

    // Gather x_src: lane d holds x[d] (one coalesced 128B load; table is L2-hot)
    const float xv = node_states[(size_t)src * NODE_DIM + lane];

    const int hi  = (lane >= 16) ? 1 : 0;   // upper half-wave flag
    const int col = lane & 15;

    // Base pointer already folded with this lane's (row-pair, column) offset;
    // all 32 matrix loads below are base + compile-time immediate offsets.
    const float* __restrict__ Abase =
        a_in + (size_t)e * (NODE_DIM * NODE_DIM) + (hi ? 2 * NODE_DIM : 0) + col;

    v8f c0 = {};   // features  0..15
    v8f c1 = {};   // features 16..31

#pragma unroll
    for (int kb = 0; kb < 8; ++kb) {
        // A operand: broadcast x[4kb + ...] to the whole half-wave (ds_bpermute)
        v2f a;
        a.x = __shfl(xv, 4 * kb + 2 * hi,     32);
        a.y = __shfl(xv, 4 * kb + 2 * hi + 1, 32);

        // B operand, feature half 0 — nontemporal: a_in is a 2GB single-use stream
        v2f b0;
        b0.x = __builtin_nontemporal_load(Abase + kb * 128 + 0);
        b0.y = __builtin_nontemporal_load(Abase + kb * 128 + 32);
        c0 = __builtin_amdgcn_wmma_f32_16x16x4_f32(false, a, false, b0,
                                                   (short)0, c0, false, false);

        // B operand, feature half 1
        v2f b1;
        b1.x = __builtin_nontemporal_load(Abase + kb * 128 + 16);
        b1.y = __builtin_nontemporal_load(Abase + kb * 128 + 48);
        c1 = __builtin_amdgcn_wmma_f32_16x16x4_f32(false, a, false, b1,
                                                   (short)0, c1, false, false);
    }

    // Every lane of c0/c1 element 0 holds msg[half*16 + col]; pick own feature.
    float val = hi ? c1[0] : c0[0];
    unsafeAtomicAdd(&accum[(size_t)dst * NODE_DIM + lane], val);
    if (lane == 0) unsafeAtomicAdd(&cnt[dst], 1.0f);
}

// ---------------------------------------------------------------------------
// Kernel 3: in-place mean + bias + ReLU over the accumulator.
// ---------------------------------------------------------------------------
__global__ void MP_finalize_kernel(float* __restrict__ out,
                                   const float* __restrict__ cnt,
                                   const float* __restrict__ bias) {
    int i = blockIdx.x * blockDim.x + threadIdx.x;
    if (i >= N_NODES * NODE_DIM) return;
    int node = i >> 5;
    int f    = i & 31;
    float c  = cnt[node];
    float m  = out[i] / fmaxf(c, 1.0f);
    out[i]   = fmaxf(m + bias[f], 0.0f);
}

// ---------------------------------------------------------------------------
extern "C" void kernel_launch(void* const* d_in, const int* in_sizes, int n_in,
                              void* d_out, int out_size, void* d_ws, size_t ws_size,
                              hipStream_t stream) {
    const float* node_states = (const float*)d_in[0];
    const int*   edge_index  = (const int*)d_in[1];
    const float* a_in        = (const float*)d_in[2];
    const float* bias        = (const float*)d_in[3];
    float* out = (float*)d_out;
    float* cnt = (float*)d_ws;          // 100000 floats = 400 KB of scratch

    const int zt = N_NODES * NODE_DIM;
    MP_zero_kernel<<<(zt + 255) / 256, 256, 0, stream>>>(out, cnt);

    const int edges_per_block = 256 / 32;   // 8 waves/block, 1 edge/wave
    MP_edge_kernel<<<(N_EDGES + edges_per_block - 1) / edges_per_block, 256, 0, stream>>>(
        node_states, edge_index, a_in, out, cnt);

    MP_finalize_kernel<<<(zt + 255) / 256, 256, 0, stream>>>(out, cnt, bias);
}